// PANLoss_87806311400038
// MI455X (gfx1250) — compile-verified
//
#include <hip/hip_runtime.h>
#include <stdint.h>

// ---------------------------------------------------------------------------
// PANLoss for MI455X (gfx1250).
// HBM-bound streaming reduction: 262 MB read -> ~11 us at 23.3 TB/s.
// CDNA5 paths exercised: GLOBAL_LOAD_ASYNC_TO_LDS_B128 double-buffered
// prefetch pipeline + S_WAIT_ASYNCCNT split counter (ASYNCcnt).
// No matmul structure -> WMMA not applicable; goal is VALU << memory time.
// ---------------------------------------------------------------------------

#define B_IMG 16
#define CCH   4
#define HW    409600                      // 640*640
#define TP    512                         // tile pixels
#define TILES_PER_IMG   (HW / TP)         // 800
#define BLOCKS_PER_IMG  50
#define TILES_PER_BLOCK (TILES_PER_IMG / BLOCKS_PER_IMG)  // 16
#define NTHR  256
#define BUF_U32 (10 * TP)                 // 5120 dwords = 20KB per buffer
#define EPSF 1e-5f
#define NACC 46                           // 6 dice + 8 ct + 8 ck + 8 ST + 8 SK + 8 SS

__device__ __forceinline__ void async_b128(uint32_t lds_byte, uint64_t gaddr) {
  // GV mode: per-lane 64-bit global address, per-lane 32-bit LDS dest.
  asm volatile("global_load_async_to_lds_b128 %0, %1, off"
               :: "v"(lds_byte), "v"(gaddr) : "memory");
}

__device__ __forceinline__ float wave_red(float v) {
  #pragma unroll
  for (int off = 16; off > 0; off >>= 1) v += __shfl_down(v, off, 32);
  return v;
}

__global__ __launch_bounds__(NTHR) void pan_main(
    const float* __restrict__ pr, const float* __restrict__ rg,
    const float* __restrict__ pk, const float* __restrict__ kg,
    const float* __restrict__ sim,
    const int* __restrict__ tl, const int* __restrict__ kl,
    float* __restrict__ ws)
{
  __shared__ uint32_t lds[2 * BUF_U32];   // 40KB static LDS (double buffer)
  const int t   = threadIdx.x;
  const int b   = blockIdx.x / BLOCKS_PER_IMG;
  const int blk = blockIdx.x % BLOCKS_PER_IMG;
  const uint32_t ldsBase = (uint32_t)(uintptr_t)(&lds[0]);

  // float accumulators (wave-reduced at the end)
  float dice[6];
  float fT[8], fK[8], fS[8];
  #pragma unroll
  for (int j = 0; j < 6; ++j) dice[j] = 0.f;
  #pragma unroll
  for (int j = 0; j < 8; ++j) { fT[j] = 0.f; fK[j] = 0.f; fS[j] = 0.f; }
  // wave-uniform counters via ballot/popc (scalar-side adds, no reduction)
  unsigned ct8[8], ck8[8];
  #pragma unroll
  for (int j = 0; j < 8; ++j) { ct8[j] = 0u; ck8[j] = 0u; }

  // per-thread copy assignment for the 6 non-sim streams
  const int hi  = (t >> 7) & 1;           // threads 128..255 take odd streams
  const int pxc = (t & 127) * 4;

  auto prefetch = [&](int tt, int bsel) {
    const int tile0 = tt * TP;
    const uint32_t bufByte = ldsBase + (uint32_t)bsel * (BUF_U32 * 4u);
    // sim: 4ch x 512 px = 512 chunks of 16B; thread handles chunk t, t+256
    #pragma unroll
    for (int u = 0; u < 2; ++u) {
      int chunk = t + u * 256;
      int c  = chunk >> 7;                // 128 chunks per channel
      int px = (chunk & 127) * 4;
      uint64_t g = (uint64_t)(uintptr_t)(sim + (size_t)(b * CCH + c) * HW + tile0 + px);
      async_b128(bufByte + (uint32_t)(c * TP + px) * 4u, g);
    }
    // streams 0..5: pr, rg, pk, kg, tl, kl at float offsets (4+s)*TP
    {
      const float* s0 = hi ? rg : pr;
      uint64_t g = (uint64_t)(uintptr_t)(s0 + (size_t)b * HW + tile0 + pxc);
      async_b128(bufByte + (uint32_t)(CCH * TP + hi * TP + pxc) * 4u, g);
    }
    {
      const float* s1 = hi ? kg : pk;
      uint64_t g = (uint64_t)(uintptr_t)(s1 + (size_t)b * HW + tile0 + pxc);
      async_b128(bufByte + (uint32_t)(CCH * TP + (2 + hi) * TP + pxc) * 4u, g);
    }
    {
      const int* s2p = hi ? kl : tl;
      uint64_t g = (uint64_t)(uintptr_t)(s2p + (size_t)b * HW + tile0 + pxc);
      async_b128(bufByte + (uint32_t)(CCH * TP + (4 + hi) * TP + pxc) * 4u, g);
    }
  };

  prefetch(blk, 0);   // tile j=0

  for (int j = 0; j < TILES_PER_BLOCK; ++j) {
    const int cur = j & 1;
    if (j + 1 < TILES_PER_BLOCK) {
      prefetch(blk + (j + 1) * BLOCKS_PER_IMG, cur ^ 1);
      // 5 async ops outstanding for next tile; wait for current tile's 5.
      asm volatile("s_wait_asynccnt 5" ::: "memory");
    } else {
      asm volatile("s_wait_asynccnt 0" ::: "memory");
    }
    __syncthreads();   // all waves' copies for this tile landed in LDS

    const uint32_t* S = &lds[cur * BUF_U32];
    #pragma unroll
    for (int u = 0; u < 2; ++u) {
      const int px = t + u * 256;        // conflict-free lane-stride-1 reads
      float s2 = 0.f;
      #pragma unroll
      for (int c = 0; c < CCH; ++c) {
        float v = __uint_as_float(S[c * TP + px]);
        s2 = fmaf(v, v, s2);
      }
      float prv = __uint_as_float(S[4 * TP + px]);
      float grv = __uint_as_float(S[5 * TP + px]);
      float pkv = __uint_as_float(S[6 * TP + px]);
      float gkv = __uint_as_float(S[7 * TP + px]);
      int   tlv = (int)S[8 * TP + px];
      int   klv = (int)S[9 * TP + px];

      float spr = 1.f / (1.f + __expf(-prv));
      float spk = 1.f / (1.f + __expf(-pkv));
      dice[0] += spr * grv;  dice[1] += spr * spr;  dice[2] += grv * grv;
      dice[3] += spk * gkv;  dice[4] += spk * spk;  dice[5] += gkv * gkv;

      #pragma unroll
      for (int i = 1; i <= 8; ++i) {
        bool mt = (tlv == i), mk = (klv == i);
        // cardinalities: wave-wide popcount (scalar path, wave-uniform)
        ct8[i - 1] += (unsigned)__popcll(__ballot(mt));
        ck8[i - 1] += (unsigned)__popcll(__ballot(mk));
        // masked feature-norm sums (per-lane, reduced later)
        fT[i - 1] += mt ? s2 : 0.f;             // ΣT
        fK[i - 1] += mk ? s2 : 0.f;             // ΣK
        fS[i - 1] += (mt && mk) ? s2 : 0.f;     // ΣS (t==k==i)
      }
    }
    __syncthreads();   // protect buf[cur] from being overwritten by prefetch
  }

  // ---- block reduction: wave shfl -> LDS staging -> per-block partials ----
  // per-wave layout: [0..5] dice, [6..13] ct, [14..21] ck,
  //                  [22..29] ST, [30..37] SK, [38..45] SS
  const int lane = t & 31;
  const int wv   = t >> 5;
  float* red = (float*)lds;              // safe: all async done, post-barrier
  #pragma unroll
  for (int jj = 0; jj < 6; ++jj) {
    float v = wave_red(dice[jj]);
    if (lane == 0) red[wv * NACC + jj] = v;
  }
  if (lane == 0) {
    #pragma unroll
    for (int i = 0; i < 8; ++i) {
      red[wv * NACC +  6 + i] = (float)ct8[i];   // already wave totals
      red[wv * NACC + 14 + i] = (float)ck8[i];
    }
  }
  #pragma unroll
  for (int i = 0; i < 8; ++i) {
    float vT = wave_red(fT[i]);
    float vK = wave_red(fK[i]);
    float vS = wave_red(fS[i]);
    if (lane == 0) {
      red[wv * NACC + 22 + i] = vT;
      red[wv * NACC + 30 + i] = vK;
      red[wv * NACC + 38 + i] = vS;
    }
  }
  __syncthreads();
  if (t < NACC) {
    float s = 0.f;
    #pragma unroll
    for (int w = 0; w < NTHR / 32; ++w) s += red[w * NACC + t];
    ws[(size_t)blockIdx.x * NACC + t] = s;   // deterministic, no atomics
  }
}

__global__ void pan_final(const float* __restrict__ ws, float* __restrict__ out) {
  const int lane = threadIdx.x;
  float lr = 0.f, lk = 0.f, lagg = 0.f, ldis = 0.f;
  if (lane < B_IMG) {
    const int b = lane;
    float A[NACC];
    #pragma unroll
    for (int j = 0; j < NACC; ++j) A[j] = 0.f;
    for (int blk = 0; blk < BLOCKS_PER_IMG; ++blk) {
      const float* p = &ws[(size_t)(b * BLOCKS_PER_IMG + blk) * NACC];
      #pragma unroll
      for (int j = 0; j < NACC; ++j) A[j] += p[j];
    }
    // dice (per-sample, summed over batch by the wave reduction below)
    float dice_r = (2.f * A[0] + EPSF) / ((A[1] + EPSF) + (A[2] + EPSF));
    float dice_k = (2.f * A[3] + EPSF) / ((A[4] + EPSF) + (A[5] + EPSF));
    lr = 1.f - dice_r;
    lk = 1.f - dice_k;
    // aggregation + discrimination
    float a[8];
    #pragma unroll
    for (int i = 0; i < 8; ++i) {
      float ct = A[ 6 + i], ck = A[14 + i];
      float ST = A[22 + i], SK = A[30 + i], SS = A[38 + i];
      float inv = 1.f / (ck + 1.f);
      float n2  = (1.f - inv) * (1.f - inv) * SS + (ST - SS) + inv * inv * (SK - SS);
      float nrm = sqrtf(n2);
      float dd  = nrm - 0.5f;                         // SIGMA_AGG
      lagg += logf(dd * dd + 1.f) / (ct + 1.f);
      float ckd = ck + 0.001f;
      a[i] = SK / (ckd * ckd);
    }
    float dsum = 0.f;
    #pragma unroll
    for (int i = 0; i < 8; ++i)
      #pragma unroll
      for (int jq = i + 1; jq < 8; ++jq) {
        float p = 3.0f - sqrtf(a[i] + a[jq]);         // SIGMA_DIS
        dsum += logf(p * p + 1.f);
      }
    ldis = dsum / 56.f;                               // K_MAX*(K_MAX-1)
  }
  lr   = wave_red(lr);
  lk   = wave_red(lk);
  lagg = wave_red(lagg);
  ldis = wave_red(ldis);
  if (lane == 0) {
    out[1] = lr;
    out[2] = lk;
    out[3] = lagg;
    out[4] = ldis;
    out[0] = lr + 0.5f * lk + 0.25f * (lagg + ldis);  // ALPHA, BETA
  }
}

extern "C" void kernel_launch(void* const* d_in, const int* in_sizes, int n_in,
                              void* d_out, int out_size, void* d_ws, size_t ws_size,
                              hipStream_t stream) {
  const float* pred_regions = (const float*)d_in[0];
  const float* regions_gt   = (const float*)d_in[1];
  const float* pred_kernels = (const float*)d_in[2];
  const float* kernels_gt   = (const float*)d_in[3];
  const float* pred_sims    = (const float*)d_in[4];
  const int*   tlab         = (const int*)d_in[5];
  const int*   klab         = (const int*)d_in[6];
  float* ws  = (float*)d_ws;    // 800 blocks * 46 floats = 147.2 KB partials
  float* out = (float*)d_out;   // 5 scalars

  pan_main<<<dim3(B_IMG * BLOCKS_PER_IMG), dim3(NTHR), 0, stream>>>(
      pred_regions, regions_gt, pred_kernels, kernels_gt,
      pred_sims, tlab, klab, ws);
  pan_final<<<dim3(1), dim3(32), 0, stream>>>(ws, out);
}